// HierarchicalGATLayer_46677704573243
// MI455X (gfx1250) — compile-verified
//
#include <hip/hip_runtime.h>
#include <math.h>

#define N_NODES 50000
#define N_EDGES 800000
#define DIM 128
#define HEADS 4
#define NEG_SLOPE 0.2f
#define LN_EPS 1e-5f

typedef __attribute__((ext_vector_type(2))) float v2f;
typedef __attribute__((ext_vector_type(8))) float v8f;

// ---------------------------------------------------------------- init
__global__ void gat_init_kernel(float* __restrict__ out, float* __restrict__ mbuf,
                                float* __restrict__ denom) {
    int i = blockIdx.x * blockDim.x + threadIdx.x;
    if (i < N_NODES * DIM) out[i] = 0.0f;
    if (i < N_NODES * HEADS) {
        mbuf[i]  = -INFINITY;
        denom[i] = 0.0f;
    }
}

// ---------------------------------------------------------------- GEMM (WMMA fp32)
// xl = x @ W_l + b_l ; xr = x @ W_r + b_r
// Block: 256 thr = 8 waves. Block b -> rows [16b,16b+16); wave w -> cols [16w,16w+16).
// fp32 WMMA 16x16x4: A frag = 2 VGPRs (lane half 0: K+0/K+1, half 1: K+2/K+3),
// B frag = 2 VGPRs (same K split, N striped over 16 lanes), C = 8 VGPRs (M = r + 8*half).
__global__ void gat_gemm_kernel(const float* __restrict__ x,
                                const float* __restrict__ Wl, const float* __restrict__ bl,
                                const float* __restrict__ Wr, const float* __restrict__ br,
                                float* __restrict__ xl, float* __restrict__ xr) {
    const int lane = threadIdx.x & 31;
    const int wave = threadIdx.x >> 5;
    const int half = lane >> 4;
    const int lm   = lane & 15;
    const int row0 = blockIdx.x * 16;
    const int n0   = wave * 16;

    v8f accL = {};
    v8f accR = {};
    const float* xrow = x + (size_t)(row0 + lm) * DIM;

#pragma unroll 4
    for (int kk = 0; kk < DIM / 4; ++kk) {
        const int k0 = kk * 4 + half * 2;
        v2f a, bL, bR;
        a.x  = xrow[k0];
        a.y  = xrow[k0 + 1];
        bL.x = Wl[(size_t)k0 * DIM + n0 + lm];
        bL.y = Wl[(size_t)(k0 + 1) * DIM + n0 + lm];
        bR.x = Wr[(size_t)k0 * DIM + n0 + lm];
        bR.y = Wr[(size_t)(k0 + 1) * DIM + n0 + lm];
        accL = __builtin_amdgcn_wmma_f32_16x16x4_f32(false, a, false, bL, (short)0, accL, false, false);
        accR = __builtin_amdgcn_wmma_f32_16x16x4_f32(false, a, false, bR, (short)0, accR, false, false);
    }

    const float biasL = bl[n0 + lm];
    const float biasR = br[n0 + lm];
#pragma unroll
    for (int r = 0; r < 8; ++r) {
        const int m = row0 + r + 8 * half;
        xl[(size_t)m * DIM + n0 + lm] = accL[r] + biasL;
        xr[(size_t)m * DIM + n0 + lm] = accR[r] + biasR;
    }
}

// ---------------------------------------------------------------- float atomic max (int trick)
__device__ __forceinline__ void atomicMaxFloat(float* addr, float val) {
    if (val >= 0.0f) {
        atomicMax((int*)addr, __float_as_int(val));
    } else {
        atomicMin((unsigned int*)addr, __float_as_uint(val));
    }
}

// ---------------------------------------------------------------- edge logits + segment max
// 1 wave per edge. Lane owns dims [4*lane, 4*lane+4); head = lane>>3.
__global__ void gat_edge_logits_kernel(const int* __restrict__ ei,
                                       const float* __restrict__ xl,
                                       const float* __restrict__ xr,
                                       const float* __restrict__ att,
                                       float* __restrict__ e,
                                       float* __restrict__ mbuf) {
    const int lane = threadIdx.x & 31;
    const int edge = blockIdx.x * 8 + (threadIdx.x >> 5);
    if (edge >= N_EDGES) return;

    const int src = ei[edge];
    const int dst = ei[N_EDGES + edge];

    const float4 a  = *(const float4*)(xl + (size_t)src * DIM + lane * 4);
    const float4 b  = *(const float4*)(xr + (size_t)dst * DIM + lane * 4);
    const float4 av = *(const float4*)(att + lane * 4);

    float s = 0.0f, t;
    t = a.x + b.x; s += (t > 0.0f ? t : NEG_SLOPE * t) * av.x;
    t = a.y + b.y; s += (t > 0.0f ? t : NEG_SLOPE * t) * av.y;
    t = a.z + b.z; s += (t > 0.0f ? t : NEG_SLOPE * t) * av.z;
    t = a.w + b.w; s += (t > 0.0f ? t : NEG_SLOPE * t) * av.w;

    // reduce across the 8 lanes of this head
    s += __shfl_xor(s, 1, 32);
    s += __shfl_xor(s, 2, 32);
    s += __shfl_xor(s, 4, 32);

    if ((lane & 7) == 0) {
        const int h = lane >> 3;
        e[(size_t)edge * HEADS + h] = s;
        atomicMaxFloat(mbuf + (size_t)dst * HEADS + h, s);
    }
}

// ---------------------------------------------------------------- exp + segment sum
__global__ void gat_edge_exp_kernel(const int* __restrict__ ei,
                                    float* __restrict__ e,          // in: logits, out: exp
                                    const float* __restrict__ mbuf,
                                    float* __restrict__ denom) {
    const int idx = blockIdx.x * blockDim.x + threadIdx.x;
    if (idx >= N_EDGES * HEADS) return;
    const int edge = idx >> 2;
    const int h    = idx & 3;
    const int dst  = ei[N_EDGES + edge];
    const float ex = expf(e[idx] - mbuf[(size_t)dst * HEADS + h]);
    e[idx] = ex;
    atomicAdd(denom + (size_t)dst * HEADS + h, ex);
}

// ---------------------------------------------------------------- weighted scatter-add
// 1 wave per edge; lane owns 4 dims; alpha per head.
__global__ void gat_scatter_kernel(const int* __restrict__ ei,
                                   const float* __restrict__ xl,
                                   const float* __restrict__ ex,
                                   const float* __restrict__ denom,
                                   float* __restrict__ out) {
    const int lane = threadIdx.x & 31;
    const int edge = blockIdx.x * 8 + (threadIdx.x >> 5);
    if (edge >= N_EDGES) return;

    const int src = ei[edge];
    const int dst = ei[N_EDGES + edge];
    const int h   = lane >> 3;

    const float alpha = ex[(size_t)edge * HEADS + h] /
                        (denom[(size_t)dst * HEADS + h] + 1e-16f);

    const float4 a = *(const float4*)(xl + (size_t)src * DIM + lane * 4);
    float* o = out + (size_t)dst * DIM + lane * 4;
    atomicAdd(o + 0, alpha * a.x);
    atomicAdd(o + 1, alpha * a.y);
    atomicAdd(o + 2, alpha * a.z);
    atomicAdd(o + 3, alpha * a.w);
}

// ---------------------------------------------------------------- bias + residual + LayerNorm
// 1 wave per node, in-place on d_out.
__global__ void gat_finalize_kernel(const float* __restrict__ x,
                                    const float* __restrict__ bias,
                                    const float* __restrict__ gamma,
                                    const float* __restrict__ beta,
                                    float* __restrict__ out) {
    const int lane = threadIdx.x & 31;
    const int node = blockIdx.x * 8 + (threadIdx.x >> 5);
    if (node >= N_NODES) return;

    float* orow = out + (size_t)node * DIM + lane * 4;
    const float4 acc = *(const float4*)orow;
    const float4 xv  = *(const float4*)(x + (size_t)node * DIM + lane * 4);
    const float4 bv  = *(const float4*)(bias + lane * 4);

    float4 v;
    v.x = acc.x + bv.x + xv.x;
    v.y = acc.y + bv.y + xv.y;
    v.z = acc.z + bv.z + xv.z;
    v.w = acc.w + bv.w + xv.w;

    float sum = v.x + v.y + v.z + v.w;
    float sq  = v.x * v.x + v.y * v.y + v.z * v.z + v.w * v.w;
#pragma unroll
    for (int m = 1; m < 32; m <<= 1) {
        sum += __shfl_xor(sum, m, 32);
        sq  += __shfl_xor(sq, m, 32);
    }
    const float mu   = sum * (1.0f / DIM);
    const float var  = sq * (1.0f / DIM) - mu * mu;
    const float rstd = rsqrtf(var + LN_EPS);

    const float4 g  = *(const float4*)(gamma + lane * 4);
    const float4 bt = *(const float4*)(beta + lane * 4);

    float4 r;
    r.x = (v.x - mu) * rstd * g.x + bt.x;
    r.y = (v.y - mu) * rstd * g.y + bt.y;
    r.z = (v.z - mu) * rstd * g.z + bt.z;
    r.w = (v.w - mu) * rstd * g.w + bt.w;
    *(float4*)orow = r;
}

// ---------------------------------------------------------------- launch
extern "C" void kernel_launch(void* const* d_in, const int* in_sizes, int n_in,
                              void* d_out, int out_size, void* d_ws, size_t ws_size,
                              hipStream_t stream) {
    const float* x     = (const float*)d_in[0];
    const int*   ei    = (const int*)d_in[1];   // [2, E] flattened: src row then dst row
    const float* W_l   = (const float*)d_in[2];
    const float* b_l   = (const float*)d_in[3];
    const float* W_r   = (const float*)d_in[4];
    const float* b_r   = (const float*)d_in[5];
    const float* att   = (const float*)d_in[6];
    const float* bias  = (const float*)d_in[7];
    const float* gamma = (const float*)d_in[8];
    const float* beta  = (const float*)d_in[9];
    float* out = (float*)d_out;

    // workspace layout (floats)
    float* xl    = (float*)d_ws;                     // N*DIM
    float* xr    = xl + (size_t)N_NODES * DIM;       // N*DIM
    float* e     = xr + (size_t)N_NODES * DIM;       // E*HEADS (logits -> exp)
    float* mbuf  = e + (size_t)N_EDGES * HEADS;      // N*HEADS
    float* denom = mbuf + (size_t)N_NODES * HEADS;   // N*HEADS

    // 1) init accumulators / segment buffers
    {
        const int total = N_NODES * DIM;
        gat_init_kernel<<<(total + 255) / 256, 256, 0, stream>>>(out, mbuf, denom);
    }
    // 2) xl / xr GEMMs (WMMA fp32)
    gat_gemm_kernel<<<N_NODES / 16, 256, 0, stream>>>(x, W_l, b_l, W_r, b_r, xl, xr);
    // 3) per-edge attention logits + segment max
    gat_edge_logits_kernel<<<(N_EDGES + 7) / 8, 256, 0, stream>>>(ei, xl, xr, att, e, mbuf);
    // 4) exp + segment sum
    gat_edge_exp_kernel<<<(N_EDGES * HEADS + 255) / 256, 256, 0, stream>>>(ei, e, mbuf, denom);
    // 5) alpha-weighted scatter-add of xl[src] into out[dst]
    gat_scatter_kernel<<<(N_EDGES + 7) / 8, 256, 0, stream>>>(ei, xl, e, denom, out);
    // 6) bias + residual + LayerNorm (in place)
    gat_finalize_kernel<<<(N_NODES + 7) / 8, 256, 0, stream>>>(x, bias, gamma, beta, out);
}